// FirstCALMEncoderLayer_35983236006604
// MI455X (gfx1250) — compile-verified
//
#include <hip/hip_runtime.h>
#include <hip/hip_bf16.h>
#include <math.h>

// ---------------------------------------------------------------------------
// Problem constants (from the reference)
// ---------------------------------------------------------------------------
#define BB   8
#define NN   2048
#define CC   32
#define DD   64
#define QQ   256
#define KSEL 205           // floor(0.1*(N-1)) + 1
#define KPAD 208           // padded to 13 * 16
#define EPSV 1e-8f

typedef float v2f __attribute__((ext_vector_type(2)));
typedef float v8f __attribute__((ext_vector_type(8)));

// CDNA5 async global->LDS path (ASYNCcnt), if the toolchain exposes it.
#if defined(__has_builtin)
#if __has_builtin(__builtin_amdgcn_global_load_async_to_lds_b32)
#define USE_ASYNC 1
#else
#define USE_ASYNC 0
#endif
#else
#define USE_ASYNC 0
#endif

typedef __attribute__((address_space(1))) int gint;   // global (AS1) int
typedef __attribute__((address_space(3))) int lint;   // LDS (AS3) int

__device__ __forceinline__ void wait_async_all() {
#if USE_ASYNC
#if __has_builtin(__builtin_amdgcn_s_wait_asynccnt)
    __builtin_amdgcn_s_wait_asynccnt(0);
#else
    asm volatile("s_wait_asynccnt 0x0" ::: "memory");
#endif
#endif
}

// fp32 WMMA: D(16x16) = A(16x4) * B(4x16) + C   -- exact fp32 semantics
__device__ __forceinline__ v8f wmma4(v2f a, v2f b, v8f c) {
    return __builtin_amdgcn_wmma_f32_16x16x4_f32(
        /*neg_a=*/false, a, /*neg_b=*/false, b,
        /*c_mod=*/(short)0, c, /*reuse_a=*/false, /*reuse_b=*/false);
}

__device__ __forceinline__ float wrapd(float x) {
    // (x + 0.5) mod 1.0 - 0.5  (jnp mod semantics: result in [0,1))
    float r = x + 0.5f;
    r -= floorf(r);
    return r - 0.5f;
}

__device__ __forceinline__ float geluf(float x) {
    // exact gelu: 0.5*x*(1+erf(x/sqrt(2)))
    return 0.5f * x * (1.0f + erff(x * 0.70710678118654752f));
}

// ---------------------------------------------------------------------------
// Kernel 0: xl[b,n,c] = x[b,n,:] @ W_lin[:,c] + b_lin[c]   (B*N = 16384 rows)
// ---------------------------------------------------------------------------
__global__ void __launch_bounds__(256)
k_xl(const float* __restrict__ x, const float* __restrict__ W,
     const float* __restrict__ b, float* __restrict__ xl) {
    int id = blockIdx.x * 256 + threadIdx.x;      // 524288 outputs
    int r = id >> 5, c = id & 31;
    const float* xr = x + (size_t)r * CC;
    float s = b[c];
#pragma unroll
    for (int i = 0; i < CC; ++i) s += xr[i] * W[i * CC + c];
    xl[id] = s;
}

// ---------------------------------------------------------------------------
// Kernel 2: per-query — kNN select, softmax weights, RFF-MLP kernel factors,
//           double-buffered async Z gather + WMMA GEMM M = G'^T * Z,
//           then WMMA combine with [W2;filt].  256 threads = 8 wave32.
// ---------------------------------------------------------------------------
__global__ void __launch_bounds__(256)
k_query(const float* __restrict__ pos,  const float* __restrict__ qp,
        const float* __restrict__ qmw,  const float* __restrict__ qmo,
        const float* __restrict__ rffB, const float* __restrict__ W1,
        const float* __restrict__ b1,   const float* __restrict__ W2,
        const float* __restrict__ filt, const float* __restrict__ bias,
        const float* __restrict__ xl,   float* __restrict__ yact) {
    __shared__ float g_s[KPAD * 34];   // weighted gelu-MLP output g'[k][j]
    __shared__ float zm_s[8256];       // phase A: 2 x Z block [16][258]; phase B: M [32][257]
    __shared__ int   sel_i_s[KPAD];    // selected global indices
    __shared__ float sel_w_s[KPAD];    // edist -> softmax weight
    __shared__ float s_red[256];       // reductions; later s[n] = sum_k w_k Z[k][n]
    __shared__ int   s_cnt[4];

    const int t = threadIdx.x;
    const int q = blockIdx.x;
    const float qp0 = qp[q * 2 + 0], qp1 = qp[q * 2 + 1];

    // ---- Phase 1: distances (8 points per thread, kept in registers) ----
    unsigned u[8]; float ev[8];
#pragma unroll
    for (int jj = 0; jj < 8; ++jj) {
        int i = t + 256 * jj;
        float d0 = wrapd(qp0 - pos[i * 2 + 0]);
        float d1 = wrapd(qp1 - pos[i * 2 + 1]);
        float e = d0 * d0 + d1 * d1;               // >= 0 -> uint order == float order
        ev[jj] = e;
        u[jj] = __float_as_uint(e);
    }

    // ---- Phase 2: binary search for the K-th smallest threshold T ----
    unsigned lo = 0u, hi = 0xFFFFFFFFu;
    for (int iter = 0; iter < 32; ++iter) {
        unsigned mid = lo + ((hi - lo) >> 1);
        if (t == 0) s_cnt[0] = 0;
        __syncthreads();
        int loc = 0;
#pragma unroll
        for (int jj = 0; jj < 8; ++jj) loc += (u[jj] <= mid) ? 1 : 0;
        atomicAdd(&s_cnt[0], loc);
        __syncthreads();
        int c = s_cnt[0];
        __syncthreads();
        if (c >= KSEL) hi = mid; else lo = mid + 1;
    }
    const unsigned T = lo;   // smallest value with count_le >= KSEL

    // ---- Phase 3: select K indices (strict-less first, then equals) ----
    if (t == 0) { s_cnt[1] = 0; s_cnt[2] = 0; s_cnt[3] = 0; }
    __syncthreads();
    {
        int loc = 0;
#pragma unroll
        for (int jj = 0; jj < 8; ++jj) loc += (u[jj] < T) ? 1 : 0;
        atomicAdd(&s_cnt[1], loc);
    }
    __syncthreads();
    const int cl = s_cnt[1];
#pragma unroll
    for (int jj = 0; jj < 8; ++jj) {
        int i = t + 256 * jj;
        if (u[jj] < T) {
            int sl = atomicAdd(&s_cnt[2], 1);
            sel_i_s[sl] = i; sel_w_s[sl] = ev[jj];
        } else if (u[jj] == T) {
            int sl = cl + atomicAdd(&s_cnt[3], 1);
            if (sl < KSEL) { sel_i_s[sl] = i; sel_w_s[sl] = ev[jj]; }
        }
    }
    __syncthreads();

    // ---- Phase 4: normalize ed, softmax weights ----
    float e = (t < KSEL) ? sel_w_s[t] : 0.f;
    s_red[t] = (t < KSEL) ? e : 3.0e38f;
    __syncthreads();
    for (int off = 128; off; off >>= 1) {
        if (t < off) s_red[t] = fminf(s_red[t], s_red[t + off]);
        __syncthreads();
    }
    float mn = s_red[0]; __syncthreads();
    s_red[t] = (t < KSEL) ? (e - mn) : -3.0e38f;
    __syncthreads();
    for (int off = 128; off; off >>= 1) {
        if (t < off) s_red[t] = fmaxf(s_red[t], s_red[t + off]);
        __syncthreads();
    }
    float mx = s_red[0]; __syncthreads();
    float wgt = (t < KSEL) ? expf(-(e - mn) / (mx + EPSV)) : 0.f;
    s_red[t] = wgt;
    __syncthreads();
    for (int off = 128; off; off >>= 1) {
        if (t < off) s_red[t] += s_red[t + off];
        __syncthreads();
    }
    float Ssum = s_red[0]; __syncthreads();
    if (t < KPAD) {
        sel_w_s[t] = (t < KSEL) ? (wgt / Ssum) : 0.f;
        if (t >= KSEL) sel_i_s[t] = 0;        // padded rows: weight 0
    }
    __syncthreads();

    // ---- Phase 5: g'[k][j] = w_k * gelu((feat_k @ W1 + b1)*qmw + qmo) ----
    if (t < KPAD) {
        int gi = sel_i_s[t];
        float wk = sel_w_s[t];
        float d0 = wrapd(qp0 - pos[gi * 2 + 0]);
        float d1 = wrapd(qp1 - pos[gi * 2 + 1]);
        float ft[32];
#pragma unroll
        for (int j = 0; j < 16; ++j) {
            float p = 6.283185307179586f * (d0 * rffB[j] + d1 * rffB[16 + j]);
            ft[j] = sinf(p);
            ft[16 + j] = cosf(p);
        }
        for (int j2 = 0; j2 < 32; ++j2) {
            float h = b1[j2];
#pragma unroll
            for (int f = 0; f < 32; ++f) h += ft[f] * W1[f * 32 + j2];
            h = h * qmw[q * 32 + j2] + qmo[q * 32 + j2];
            g_s[t * 34 + j2] = wk * geluf(h);
        }
    }
    __syncthreads();

    // ---- Phase 6: M(32x256) = G'^T * Z, double-buffered async Z gather ----
    const int lane = t & 31;
    const int w    = t >> 5;            // wave id 0..7
    const int ln   = lane & 15;
    const int koff = (lane >= 16) ? 2 : 0;
    const int jb   = (w & 1) * 16;      // M-tile (rows j)
    const int nbase = (w >> 1) * 64;    // 4 N-tiles per wave
    const int bq = t >> 5, ccol = t & 31;   // thread t owns Z column n = t

    float* const zbuf0 = zm_s;          // [16][258]
    float* const zbuf1 = zm_s + 4128;   // [16][258]

    auto prefetchZ = [&](int blk2, float* zb) {
        const int kb2 = blk2 * 16;
#pragma unroll
        for (int kk = 0; kk < 16; ++kk) {
            int gi = sel_i_s[kb2 + kk];
            const float* gp = xl + ((size_t)bq * NN + gi) * CC + ccol;
            float* lp = &zb[kk * 258 + t];
#if USE_ASYNC
            __builtin_amdgcn_global_load_async_to_lds_b32(
                (gint*)(uintptr_t)gp, (lint*)(unsigned)(uintptr_t)lp, 0, 0);
#else
            *lp = *gp;
#endif
        }
    };

    prefetchZ(0, zbuf0);
    v8f acc[4] = {};
    float s_acc = 0.f;
    for (int blk = 0; blk < 13; ++blk) {
        wait_async_all();
        __syncthreads();                        // block blk resident, prev reads done
        float* zb = (blk & 1) ? zbuf1 : zbuf0;
        if (blk + 1 < 13) prefetchZ(blk + 1, (blk & 1) ? zbuf0 : zbuf1);
        const int kb = blk * 16;
        // weighted column sum for the filt term
#pragma unroll
        for (int kk = 0; kk < 16; ++kk)
            s_acc += sel_w_s[kb + kk] * zb[kk * 258 + t];
        // WMMA over the 16-deep block
#pragma unroll
        for (int c4 = 0; c4 < 4; ++c4) {
            int k0 = c4 * 4;
            v2f a;
            a.x = g_s[(kb + k0 + koff)     * 34 + jb + ln];
            a.y = g_s[(kb + k0 + koff + 1) * 34 + jb + ln];
#pragma unroll
            for (int tt = 0; tt < 4; ++tt) {
                int nb = nbase + tt * 16;
                v2f bf;
                bf.x = zb[(k0 + koff)     * 258 + nb + ln];
                bf.y = zb[(k0 + koff + 1) * 258 + nb + ln];
                acc[tt] = wmma4(a, bf, acc[tt]);
            }
        }
        __syncthreads();                        // all reads of zb done before overwrite
    }
    // spill M into zm_s (stride 257) and s[n] into s_red
#pragma unroll
    for (int tt = 0; tt < 4; ++tt) {
        int nb = nbase + tt * 16;
#pragma unroll
        for (int i = 0; i < 8; ++i) {
            int row = jb + i + ((lane < 16) ? 0 : 8);
            zm_s[row * 257 + nb + ln] = acc[tt][i];
        }
    }
    s_red[t] = s_acc;
    __syncthreads();

    // ---- Phase 7: Y(8x64) = A(8x1056) x [W2;filt](1056x64) via WMMA ----
    // A[b, j*32+c] = M[j, b*32+c]; A[b, 1024+c] = s[b*32+c];
    // B is W2 viewed flat as (1024 x 64) row-major, then filt as (32 x 64).
    if (w < 4) {
        const int nb = w * 16;
        v8f yacc = {};
        for (int kc = 0; kc < 256; ++kc) {
            int ka = kc * 4 + koff, kb2 = ka + 1;
            v2f a;
            a.x = (ln < 8) ? zm_s[(ka  >> 5) * 257 + ln * 32 + (ka  & 31)] : 0.f;
            a.y = (ln < 8) ? zm_s[(kb2 >> 5) * 257 + ln * 32 + (kb2 & 31)] : 0.f;
            v2f bf;
            bf.x = W2[ka  * 64 + nb + ln];
            bf.y = W2[kb2 * 64 + nb + ln];
            yacc = wmma4(a, bf, yacc);
        }
#pragma unroll
        for (int kc = 0; kc < 8; ++kc) {
            int ka = kc * 4 + koff, kb2 = ka + 1;
            v2f a;
            a.x = (ln < 8) ? s_red[ln * 32 + ka ] : 0.f;
            a.y = (ln < 8) ? s_red[ln * 32 + kb2] : 0.f;
            v2f bf;
            bf.x = filt[ka  * 64 + nb + ln];
            bf.y = filt[kb2 * 64 + nb + ln];
            yacc = wmma4(a, bf, yacc);
        }
        if (lane < 16) {                 // rows 0..7 (= batch b) are valid
#pragma unroll
            for (int i = 0; i < 8; ++i) {
                int d = nb + lane;
                float yv = geluf(yacc[i] + bias[d]);
                yact[((size_t)i * QQ + q) * DD + d] = yv;
            }
        }
    }
}

// ---------------------------------------------------------------------------
// Kernel 3: out = gelu(Y @ Wm1 + bm1) @ Wm2 + bm2 + Y  (rows = B*Q = 2048)
//           One 16-row tile per workgroup; WMMA f32 16x16x4 for both GEMMs.
// ---------------------------------------------------------------------------
__global__ void __launch_bounds__(256)
k_mlp(const float* __restrict__ yact, const float* __restrict__ Wm1,
      const float* __restrict__ bm1,  const float* __restrict__ Wm2,
      const float* __restrict__ bm2,  float* __restrict__ out) {
    __shared__ float y_t[16 * 68];      // Y tile (16 x 64)
    __shared__ float h_t[16 * 260];     // hidden (16 x 256)
    const int t = threadIdx.x;
    const int r0 = blockIdx.x * 16;

    for (int idx = t; idx < 16 * 64; idx += 256) {
        int r = idx >> 6, c = idx & 63;
        y_t[r * 68 + c] = yact[(size_t)(r0 + r) * 64 + c];
    }
    __syncthreads();

    const int lane = t & 31, w = t >> 5, ln = lane & 15;
    const int koff = (lane >= 16) ? 2 : 0;

    // GEMM1: H = gelu(Y(16x64) @ Wm1(64x256) + bm1); wave w -> cols [32w,32w+32)
    v8f acc[2] = {};
    for (int kc = 0; kc < 16; ++kc) {
        int k0 = kc * 4;
        v2f a;
        a.x = y_t[ln * 68 + k0 + koff];
        a.y = y_t[ln * 68 + k0 + koff + 1];
#pragma unroll
        for (int tt = 0; tt < 2; ++tt) {
            int nb = w * 32 + tt * 16;
            v2f bf;
            bf.x = Wm1[(k0 + koff)     * 256 + nb + ln];
            bf.y = Wm1[(k0 + koff + 1) * 256 + nb + ln];
            acc[tt] = wmma4(a, bf, acc[tt]);
        }
    }
#pragma unroll
    for (int tt = 0; tt < 2; ++tt) {
        int nb = w * 32 + tt * 16;
#pragma unroll
        for (int i = 0; i < 8; ++i) {
            int row = i + ((lane < 16) ? 0 : 8);
            int col = nb + ln;
            h_t[row * 260 + col] = geluf(acc[tt][i] + bm1[col]);
        }
    }
    __syncthreads();

    // GEMM2: Out(16x64) = H(16x256) @ Wm2(256x64) + bm2 + Y  (waves 0..3)
    if (w < 4) {
        int nb2 = w * 16;
        v8f a2 = {};
        for (int kc = 0; kc < 64; ++kc) {
            int k0 = kc * 4;
            v2f a;
            a.x = h_t[ln * 260 + k0 + koff];
            a.y = h_t[ln * 260 + k0 + koff + 1];
            v2f bf;
            bf.x = Wm2[(k0 + koff)     * 64 + nb2 + ln];
            bf.y = Wm2[(k0 + koff + 1) * 64 + nb2 + ln];
            a2 = wmma4(a, bf, a2);
        }
#pragma unroll
        for (int i = 0; i < 8; ++i) {
            int row = i + ((lane < 16) ? 0 : 8);
            int d = nb2 + ln;
            out[(size_t)(r0 + row) * 64 + d] = a2[i] + bm2[d] + y_t[row * 68 + d];
        }
    }
}

// ---------------------------------------------------------------------------
// Kernel 4: append query_pos (256x2) after y in d_out
// ---------------------------------------------------------------------------
__global__ void __launch_bounds__(256)
k_copyqp(const float* __restrict__ qp, float* __restrict__ out) {
    int i = blockIdx.x * 256 + threadIdx.x;
    if (i < QQ * 2) out[i] = qp[i];
}

// ---------------------------------------------------------------------------
extern "C" void kernel_launch(void* const* d_in, const int* in_sizes, int n_in,
                              void* d_out, int out_size, void* d_ws, size_t ws_size,
                              hipStream_t stream) {
    const float* x     = (const float*)d_in[0];
    const float* pos   = (const float*)d_in[1];
    const float* qp    = (const float*)d_in[2];
    const float* qmw   = (const float*)d_in[3];
    const float* qmo   = (const float*)d_in[4];
    const float* W_lin = (const float*)d_in[5];
    const float* b_lin = (const float*)d_in[6];
    const float* rffB  = (const float*)d_in[7];
    const float* W1    = (const float*)d_in[8];
    const float* b1    = (const float*)d_in[9];
    const float* W2    = (const float*)d_in[10];
    const float* filt  = (const float*)d_in[11];
    const float* bias  = (const float*)d_in[12];
    const float* Wm1   = (const float*)d_in[13];
    const float* bm1   = (const float*)d_in[14];
    const float* Wm2   = (const float*)d_in[15];
    const float* bm2   = (const float*)d_in[16];

    float* out  = (float*)d_out;
    float* xl   = (float*)d_ws;                 // B*N*C = 524288 floats
    float* yact = xl + (size_t)BB * NN * CC;    // B*Q*D = 131072 floats

    k_xl    <<<(BB * NN * CC) / 256, 256, 0, stream>>>(x, W_lin, b_lin, xl);
    k_query <<<QQ, 256, 0, stream>>>(pos, qp, qmw, qmo, rffB, W1, b1,
                                     W2, filt, bias, xl, yact);
    k_mlp   <<<(BB * QQ) / 16, 256, 0, stream>>>(yact, Wm1, bm1, Wm2, bm2, out);
    k_copyqp<<<2, 256, 0, stream>>>(qp, out + (size_t)BB * QQ * DD);
}